// DKVMN_30039001268236
// MI455X (gfx1250) — compile-verified
//
#include <hip/hip_runtime.h>
#include <hip/hip_bf16.h>

// ---------------- problem constants (from reference) ----------------
constexpr int Bn = 128;
constexpr int Tn = 200;
constexpr int BT = Bn * Tn;       // 25600 rows, divisible by 64
constexpr int Dd = 64;
constexpr int Mm = 50;
constexpr int NC = 10000;

typedef __attribute__((ext_vector_type(2))) float v2f;
typedef __attribute__((ext_vector_type(4))) float v4f;
typedef __attribute__((ext_vector_type(8))) float v8f;

// CDNA5 f32 WMMA: D(16x16,f32) = A(16x4,f32) x B(4x16,f32) + C
__device__ __forceinline__ v8f wmma_f32_4(v2f a, v2f b, v8f c) {
    return __builtin_amdgcn_wmma_f32_16x16x4_f32(
        /*neg_a=*/false, a, /*neg_b=*/false, b,
        /*c_mod=*/(short)0, c, /*reuse_a=*/false, /*reuse_b=*/false);
}

// =====================================================================
// Kernel 1: gather k/v, compute w = softmax(k*Mk^T), e = sigmoid(v*We^T+be),
//           a = tanh(v*Wa^T+ba).  One block = 64 rows, 4 waves x 16 rows.
// Weight LDS tiles are stored PRE-PAIRED along K so a B fragment is a
// single aligned 8-byte ds load: W[p][n] = { W^T[2p][n], W^T[2p+1][n] }.
// =====================================================================
__global__ __launch_bounds__(128) void k1_wea(
    const int* __restrict__ q, const int* __restrict__ r,
    const float* __restrict__ k_emb, const float* __restrict__ v_emb,
    const float* __restrict__ Mk,
    const float* __restrict__ We, const float* __restrict__ be,
    const float* __restrict__ Wa, const float* __restrict__ ba,
    float* __restrict__ w_g, float* __restrict__ e_g, float* __restrict__ a_g)
{
    __shared__ v2f   MkP[32][64];   // [kpair][n], n>=Mm zero padded
    __shared__ v2f   WeP[32][64];
    __shared__ v2f   WaP[32][64];
    __shared__ float kl[64][64];    // k rows; reused as logits scratch
    __shared__ float vl[64][64];    // v rows
    __shared__ float rmax[64], rsum[64];

    const int tid = threadIdx.x;
    const int rowbase = blockIdx.x * 64;

    // stage weights pre-paired: pair p of column n holds W[n][2p], W[n][2p+1]
    for (int i = tid; i < 32 * 64; i += 128) {
        int p = i >> 6, n = i & 63;
        if (n < Mm) MkP[p][n] = *(const v2f*)&Mk[n * Dd + 2 * p];
        else        MkP[p][n] = (v2f){0.0f, 0.0f};
        WeP[p][n] = *(const v2f*)&We[n * Dd + 2 * p];
        WaP[p][n] = *(const v2f*)&Wa[n * Dd + 2 * p];
    }
    // gather embedding rows (float4 per thread)
    for (int i = tid; i < 64 * 16; i += 128) {
        int rr = i >> 4, d4 = (i & 15) * 4;
        int row = rowbase + rr;
        int qi = q[row];
        int xi = qi + NC * r[row];
        *(v4f*)&kl[rr][d4] = *(const v4f*)&k_emb[qi * Dd + d4];
        *(v4f*)&vl[rr][d4] = *(const v4f*)&v_emb[xi * Dd + d4];
    }
    __syncthreads();

    const int wv = tid >> 5, lane = tid & 31;
    const int half = lane >> 4, lc = lane & 15;
    const int r0 = wv * 16;

    // ---- logits = k * Mk^T (16x64 per wave) ----
    v8f cw[4];
    #pragma unroll
    for (int nt = 0; nt < 4; ++nt) {
        v8f c = {};
        #pragma unroll
        for (int kk = 0; kk < 16; ++kk) {
            v2f av = *(const v2f*)&kl[r0 + lc][kk * 4 + 2 * half];
            v2f bv = MkP[kk * 2 + half][nt * 16 + lc];
            c = wmma_f32_4(av, bv, c);
        }
        cw[nt] = c;
    }
    __syncthreads();            // all waves done reading kl
    // dump logits into kl (scratch) in [row][m] layout
    #pragma unroll
    for (int nt = 0; nt < 4; ++nt)
        #pragma unroll
        for (int i = 0; i < 8; ++i)
            kl[r0 + i + 8 * half][nt * 16 + lc] = cw[nt][i];
    __syncthreads();

    // softmax stats: one thread per row (lanes 0-15 of each wave)
    if (lane < 16) {
        int rr = r0 + lc;
        float mx = -1e30f;
        for (int m = 0; m < Mm; ++m) mx = fmaxf(mx, kl[rr][m]);
        float s = 0.0f;
        for (int m = 0; m < Mm; ++m) s += __expf(kl[rr][m] - mx);
        rmax[rr] = mx;
        rsum[rr] = 1.0f / s;
    }
    __syncthreads();
    // write w (stride 64, zero padded beyond Mm)
    for (int i = tid; i < 64 * 64; i += 128) {
        int rr = i >> 6, m = i & 63;
        float val = (m < Mm) ? __expf(kl[rr][m] - rmax[rr]) * rsum[rr] : 0.0f;
        w_g[(size_t)(rowbase + rr) * 64 + m] = val;
    }

    // ---- e = sigmoid(v*We^T + be), a = tanh(v*Wa^T + ba) ----
    #pragma unroll
    for (int nt = 0; nt < 4; ++nt) {
        v8f ce = {}, ca = {};
        #pragma unroll
        for (int kk = 0; kk < 16; ++kk) {
            v2f av  = *(const v2f*)&vl[r0 + lc][kk * 4 + 2 * half];
            v2f bve = WeP[kk * 2 + half][nt * 16 + lc];
            v2f bva = WaP[kk * 2 + half][nt * 16 + lc];
            ce = wmma_f32_4(av, bve, ce);
            ca = wmma_f32_4(av, bva, ca);
        }
        int col = nt * 16 + lc;
        float bbe = be[col], bba = ba[col];
        #pragma unroll
        for (int i = 0; i < 8; ++i) {
            size_t row = (size_t)(rowbase + r0 + i + 8 * half);
            e_g[row * 64 + col] = 1.0f / (1.0f + __expf(-(ce[i] + bbe)));
            a_g[row * 64 + col] = tanhf(ca[i] + bba);
        }
    }
}

// =====================================================================
// Kernel 2: per-batch sequential scan over T.  Mv state (50x64) in LDS.
// grid = B blocks, 320 threads (10 waves): 10 state elements / thread.
// =====================================================================
__global__ __launch_bounds__(320) void k2_scan(
    const float* __restrict__ Mv0,
    const float* __restrict__ w_g, const float* __restrict__ e_g,
    const float* __restrict__ a_g, float* __restrict__ reads_g)
{
    __shared__ float Mv[Mm][64];
    __shared__ float wbuf[64], ebuf[64], abuf[64];
    __shared__ float part[5][64];

    const int tid = threadIdx.x;
    const int b = blockIdx.x;

    for (int i = tid; i < Mm * Dd; i += 320) Mv[i >> 6][i & 63] = Mv0[i];
    __syncthreads();

    for (int t = 0; t < Tn; ++t) {
        const size_t row = (size_t)b * Tn + t;
        // stage this step's w/e/a rows (float4), prefetch next step's rows
        if (tid < 16)       *(v4f*)&wbuf[tid * 4]        = *(const v4f*)&w_g[row * 64 + tid * 4];
        else if (tid < 32)  *(v4f*)&ebuf[(tid - 16) * 4] = *(const v4f*)&e_g[row * 64 + (tid - 16) * 4];
        else if (tid < 48)  *(v4f*)&abuf[(tid - 32) * 4] = *(const v4f*)&a_g[row * 64 + (tid - 32) * 4];
        else if (tid < 54 && (t + 1) < Tn) {
            int j = tid - 48;                 // 6 prefetch threads, 2 lines each buf
            const float* base = (j < 2) ? w_g : (j < 4) ? e_g : a_g;
            __builtin_prefetch(&base[(row + 1) * 64 + (j & 1) * 32], 0, 0);
        }
        __syncthreads();

        // read_t[d] = sum_m w[m]*Mv[m][d]  (5 partial rows of 10 terms)
        {
            const int m0 = tid >> 6;           // 0..4
            const int d  = tid & 63;
            float acc = 0.0f;
            #pragma unroll
            for (int j = 0; j < 10; ++j) {
                int m = m0 * 10 + j;
                acc += wbuf[m] * Mv[m][d];
            }
            part[m0][d] = acc;
        }
        __syncthreads();

        if (tid < 64) {
            float s = part[0][tid] + part[1][tid] + part[2][tid]
                    + part[3][tid] + part[4][tid];
            reads_g[row * 64 + tid] = s;
        }
        // Mv update: Mv = Mv*(1 - w⊗e) + w⊗a
        #pragma unroll
        for (int j = 0; j < 10; ++j) {
            int idx = tid + 320 * j;
            int m = idx >> 6, d = idx & 63;
            float wm = wbuf[m];
            Mv[m][d] = Mv[m][d] * (1.0f - wm * ebuf[d]) + wm * abuf[d];
        }
        __syncthreads();
    }
}

// =====================================================================
// Kernel 3: f = tanh([reads,k] * Wf^T + bf);  p = sigmoid(f . Wp + bp)
// One block = 64 rows, 4 waves x 16 rows, K=128 contraction.
// =====================================================================
__global__ __launch_bounds__(128) void k3_out(
    const int* __restrict__ q, const float* __restrict__ k_emb,
    const float* __restrict__ reads_g,
    const float* __restrict__ Wf, const float* __restrict__ bf,
    const float* __restrict__ Wp, const float* __restrict__ bp,
    float* __restrict__ out)
{
    __shared__ v2f   WfP[64][64];    // pre-paired: [kpair][n]
    __shared__ float rk[64][128];    // concat(reads, k) per row
    __shared__ float fl[64][64];
    __shared__ float wp_s[64];
    __shared__ float bp_s;

    const int tid = threadIdx.x;
    const int rowbase = blockIdx.x * 64;

    for (int i = tid; i < 64 * 64; i += 128) {
        int p = i >> 6, n = i & 63;
        WfP[p][n] = *(const v2f*)&Wf[n * 128 + 2 * p];
    }
    if (tid < 64) wp_s[tid] = Wp[tid];
    if (tid == 64) bp_s = bp[0];
    for (int i = tid; i < 64 * 32; i += 128) {
        int rr = i >> 5, c4 = (i & 31) * 4;
        size_t row = (size_t)(rowbase + rr);
        v4f val;
        if (c4 < 64) val = *(const v4f*)&reads_g[row * 64 + c4];
        else         val = *(const v4f*)&k_emb[q[row] * Dd + (c4 - 64)];
        *(v4f*)&rk[rr][c4] = val;
    }
    __syncthreads();

    const int wv = tid >> 5, lane = tid & 31;
    const int half = lane >> 4, lc = lane & 15;
    const int r0 = wv * 16;

    #pragma unroll
    for (int nt = 0; nt < 4; ++nt) {
        v8f c = {};
        #pragma unroll
        for (int kk = 0; kk < 32; ++kk) {
            v2f av = *(const v2f*)&rk[r0 + lc][kk * 4 + 2 * half];
            v2f bv = WfP[kk * 2 + half][nt * 16 + lc];
            c = wmma_f32_4(av, bv, c);
        }
        int col = nt * 16 + lc;
        float bb = bf[col];
        #pragma unroll
        for (int i = 0; i < 8; ++i)
            fl[r0 + i + 8 * half][col] = tanhf(c[i] + bb);
    }
    __syncthreads();

    if (lane < 16) {
        int rr = r0 + lc;
        float s = bp_s;
        for (int d = 0; d < 64; ++d) s += fl[rr][d] * wp_s[d];
        out[rowbase + rr] = 1.0f / (1.0f + __expf(-s));
    }
}

// =====================================================================
extern "C" void kernel_launch(void* const* d_in, const int* in_sizes, int n_in,
                              void* d_out, int out_size, void* d_ws, size_t ws_size,
                              hipStream_t stream)
{
    const int*   q     = (const int*)  d_in[0];
    const int*   r     = (const int*)  d_in[1];
    // d_in[2] = diff (unused by reference math)
    const float* k_emb = (const float*)d_in[3];
    const float* v_emb = (const float*)d_in[4];
    const float* Mk    = (const float*)d_in[5];
    const float* Mv0   = (const float*)d_in[6];
    const float* We    = (const float*)d_in[7];
    const float* be    = (const float*)d_in[8];
    const float* Wa    = (const float*)d_in[9];
    const float* ba    = (const float*)d_in[10];
    const float* Wf    = (const float*)d_in[11];
    const float* bf    = (const float*)d_in[12];
    const float* Wp    = (const float*)d_in[13];
    const float* bp    = (const float*)d_in[14];
    float* out = (float*)d_out;

    // workspace layout: w | e | a | reads, each BT x 64 f32 (~6.55 MB each)
    float* w_g     = (float*)d_ws;
    float* e_g     = w_g + (size_t)BT * 64;
    float* a_g     = e_g + (size_t)BT * 64;
    float* reads_g = a_g + (size_t)BT * 64;

    k1_wea<<<BT / 64, 128, 0, stream>>>(q, r, k_emb, v_emb, Mk, We, be, Wa, ba,
                                        w_g, e_g, a_g);
    k2_scan<<<Bn, 320, 0, stream>>>(Mv0, w_g, e_g, a_g, reads_g);
    k3_out<<<BT / 64, 128, 0, stream>>>(q, k_emb, reads_g, Wf, bf, Wp, bp, out);
}